// MatchLoss_2104533975649
// MI455X (gfx1250) — compile-verified
//
#include <hip/hip_runtime.h>
#include <stdint.h>

// ---------------------------------------------------------------------------
// MatchLoss pair-extraction for B=8192 square logits.
//   out[0..B)    = diag(logits)                  (pos, pass 1)
//   out[B..2B)   = gumbel-argmax sampled neg/row (sel, pass 1)
//   out[2B..3B)  = diag(logits)                  (pos, pass 2 = transpose)
//   out[3B..4B)  = gumbel-argmax sampled neg/col (sel, pass 2)
//   out[4B..8B)  = labels 1,0,1,0 blocks
// Memory-bound: 2 x 256MB streamed (rows then cols) ~ 22us @ 23.3 TB/s.
// Row pass streams rows HBM->LDS via the CDNA5 Tensor Data Mover (TDM),
// double-buffered with s_wait_tensorcnt. Column pass uses coalesced NT loads.
// ---------------------------------------------------------------------------

#define BDIM 8192
#define ROWS_PER_BLOCK 32
#define NEG_INF (-3.402823466e38f)
#define EPS_F 1e-4f
#define SEED1 0x9E3779B9u
#define SEED2 0x7F4A7C15u

typedef unsigned int v4u __attribute__((ext_vector_type(4)));
typedef int          v8i __attribute__((ext_vector_type(8)));
typedef int          v4i __attribute__((ext_vector_type(4)));

// Stateless Gumbel(0,1) noise from a murmur3-finalizer hash. Cheap enough
// (6 int ops + 2 fast logs) to stay under the HBM roofline, unlike threefry.
__device__ __forceinline__ float gumbel_hash(uint32_t x, uint32_t seed) {
  x ^= seed;
  x ^= x >> 16; x *= 0x85ebca6bu;
  x ^= x >> 13; x *= 0xc2b2ae35u;
  x ^= x >> 16;
  // uniform in (0,1): 24 mantissa bits, offset by 2^-25 so log() is finite
  float u = (float)(x >> 8) * 5.9604645e-08f + 2.9802322e-08f;
  return -__logf(-__logf(u));
}

// Issue a TDM DMA of one 8192-float row (32KB) into LDS.
// D# per CDNA5 ISA ch.8: group0 = {count=1 | lds_addr | global_addr | type=2},
// group1 = {data_size=4B, tensor 8192x8192 (stride 8192), tile 8192x1}.
// amdgpu-toolchain builtin arity: (v4u g0, v8i g1, v4i, v4i, v8i, i32 cpol).
__device__ __forceinline__ void tdm_load_row(const float* gsrc, uint32_t lds_off) {
  uint64_t ga = (uint64_t)(uintptr_t)gsrc;
  v4u g0;
  g0.x = 1u;                                             // count=1, user mode
  g0.y = lds_off;                                        // LDS byte address
  g0.z = (uint32_t)ga;                                   // global_addr[31:0]
  g0.w = (uint32_t)((ga >> 32) & 0x01FFFFFFu)            // global_addr[56:32]
         | 0x80000000u;                                  // type=2 ("image")
  v8i g1;
  g1[0] = 0x00020000;                                    // data_size=2 -> 4B
  g1[1] = (int)(((uint32_t)BDIM & 0xFFFFu) << 16);       // tensor_dim0 lo16
  g1[2] = (int)(((uint32_t)BDIM >> 16) |
                (((uint32_t)BDIM & 0xFFFFu) << 16));     // dim0 hi | dim1 lo
  g1[3] = (int)(((uint32_t)BDIM >> 16) |
                (((uint32_t)BDIM & 0xFFFFu) << 16));     // dim1 hi | tile_dim0
  g1[4] = 1;                                             // tile_dim1=1
  g1[5] = BDIM;                                          // dim0_stride lo32
  g1[6] = 0;
  g1[7] = 0;
  v4i z4 = {0, 0, 0, 0};                                 // groups 2/3 unused (2D)
  v8i z8 = {0, 0, 0, 0, 0, 0, 0, 0};
  __builtin_amdgcn_tensor_load_to_lds(g0, g1, z4, z4, z8, 0);
}

// ---------------- Pass 1: one block per 32 rows, TDM double-buffer ----------
__global__ __launch_bounds__(256) void row_pass_kernel(
    const float* __restrict__ logits, float* __restrict__ out) {
  __shared__ float buf[2][BDIM];                         // 2 x 32KB = 64KB LDS
  const int tid  = threadIdx.x;
  const int row0 = blockIdx.x * ROWS_PER_BLOCK;
  const bool issuer = (tid < 32);                        // wave 0 only (wave32)

  if (issuer) {
    tdm_load_row(logits + (size_t)row0 * BDIM, (uint32_t)(uintptr_t)&buf[0][0]);
  }

  for (int i = 0; i < ROWS_PER_BLOCK; ++i) {
    const int row = row0 + i;
    float* rbuf = &buf[i & 1][0];

    if (issuer) {
      if (i + 1 < ROWS_PER_BLOCK) {
        asm volatile("s_wait_dscnt 0" ::: "memory");     // LDS reads drained
        tdm_load_row(logits + (size_t)(row + 1) * BDIM,
                     (uint32_t)(uintptr_t)&buf[(i + 1) & 1][0]);
        __builtin_amdgcn_s_wait_tensorcnt(1);            // row i landed
      } else {
        __builtin_amdgcn_s_wait_tensorcnt(0);            // last row landed
      }
      asm volatile("" ::: "memory");
    }
    __syncthreads();                                     // (A) buffer ready

    float bestK = NEG_INF, bestV = 0.0f;
    #pragma unroll 8
    for (int k = 0; k < BDIM / 256; ++k) {
      const int c = (k << 8) + tid;
      const float v = rbuf[c];
      const uint32_t h = ((uint32_t)row << 13) | (uint32_t)c;  // row*8192+c
      const float key = v + (EPS_F + gumbel_hash(h, SEED1));
      if (c != row && key > bestK) { bestK = key; bestV = v; }
    }
    const float diag = rbuf[row];                        // broadcast read

    // wave32 shuffle reduction of (key, value)
    #pragma unroll
    for (int off = 16; off > 0; off >>= 1) {
      const float ok = __shfl_xor(bestK, off, 32);
      const float ov = __shfl_xor(bestV, off, 32);
      if (ok > bestK) { bestK = ok; bestV = ov; }
    }

    __syncthreads();                                     // (B) all reads done
    if ((tid & 31) == 0) {                               // reuse rbuf head
      rbuf[(tid >> 5) * 2 + 0] = bestK;
      rbuf[(tid >> 5) * 2 + 1] = bestV;
    }
    __syncthreads();                                     // (C) partials visible
    if (tid == 0) {
      float bk = rbuf[0], bv = rbuf[1];
      #pragma unroll
      for (int w = 1; w < 8; ++w) {
        const float k2 = rbuf[2 * w];
        if (k2 > bk) { bk = k2; bv = rbuf[2 * w + 1]; }
      }
      out[row]        = diag;                            // pos, pass 1
      out[BDIM + row] = bv;                              // sel, pass 1
    }
    __syncthreads();                                     // (D) safe to overwrite
  }
}

// ---------------- Pass 2: thread-per-column, coalesced NT streams -----------
__global__ __launch_bounds__(256) void col_partial_kernel(
    const float* __restrict__ logits, float2* __restrict__ part,
    int rows_per_chunk) {
  const int c  = blockIdx.x * 256 + threadIdx.x;         // column owned
  const int r0 = blockIdx.y * rows_per_chunk;
  const float* p = logits + (size_t)r0 * BDIM + c;

  float bestK = NEG_INF, bestV = 0.0f;
  for (int rr = 0; rr < rows_per_chunk; rr += 8) {
    #pragma unroll
    for (int k = 0; k < 8; ++k) {
      const int r = r0 + rr + k;
      const float v = __builtin_nontemporal_load(p + (size_t)(rr + k) * BDIM);
      const uint32_t h = ((uint32_t)c << 13) | (uint32_t)r;    // c*8192+r
      const float key = v + (EPS_F + gumbel_hash(h, SEED2));
      if (r != c && key > bestK) { bestK = key; bestV = v; }
    }
  }
  part[blockIdx.y * BDIM + c] = make_float2(bestK, bestV);
}

__global__ __launch_bounds__(256) void col_final_kernel(
    const float* __restrict__ logits, const float2* __restrict__ part,
    float* __restrict__ out, int nchunk) {
  const int c = blockIdx.x * 256 + threadIdx.x;
  float bk = NEG_INF, bv = 0.0f;
  for (int k = 0; k < nchunk; ++k) {
    const float2 pr = part[k * BDIM + c];
    if (pr.x > bk) { bk = pr.x; bv = pr.y; }
  }
  out[2 * BDIM + c] = logits[(size_t)c * (BDIM + 1)];    // pos, pass 2 (diag)
  out[3 * BDIM + c] = bv;                                // sel, pass 2
  out[4 * BDIM + c] = 1.0f;                              // labels
  out[5 * BDIM + c] = 0.0f;
  out[6 * BDIM + c] = 1.0f;
  out[7 * BDIM + c] = 0.0f;
}

extern "C" void kernel_launch(void* const* d_in, const int* in_sizes, int n_in,
                              void* d_out, int out_size, void* d_ws, size_t ws_size,
                              hipStream_t stream) {
  const float* logits = (const float*)d_in[0];
  float* out = (float*)d_out;
  float2* part = (float2*)d_ws;

  // Split the column pass over row-chunks for parallelism if scratch allows.
  const int nchunk =
      (ws_size >= (size_t)8 * BDIM * sizeof(float2)) ? 8 : 1;

  row_pass_kernel<<<BDIM / ROWS_PER_BLOCK, 256, 0, stream>>>(logits, out);
  col_partial_kernel<<<dim3(BDIM / 256, nchunk), 256, 0, stream>>>(
      logits, part, BDIM / nchunk);
  col_final_kernel<<<BDIM / 256, 256, 0, stream>>>(logits, part, out, nchunk);
}